// InvariantPointAttention_1202590843115
// MI455X (gfx1250) — compile-verified
//
#include <hip/hip_runtime.h>
#include <math.h>

// ---------------------------------------------------------------------------
// IPA forward for gfx1250 (MI455X), fp32 path using V_WMMA_F32_16X16X4_F32.
// B=1, N=768, C1=384, C2=128, H=12, SQK=SV=16, PQK=4, PV=8.
// ---------------------------------------------------------------------------

typedef __attribute__((ext_vector_type(2))) float v2f;
typedef __attribute__((ext_vector_type(8))) float v8f;

#define NN 768
#define HH 12

__device__ __forceinline__ v8f wmma4(v2f a, v2f b, v8f c) {
  // D = A(16x4 f32) * B(4x16 f32) + C(16x16 f32)
  return __builtin_amdgcn_wmma_f32_16x16x4_f32(
      false, a, false, b, (short)0, c, false, false);
}

// ---------------------------------------------------------------------------
// Generic GEMM (compile-time shapes): C[M,NOUT] = A[M,K] @ W[K,NOUT] (+bias).
// Row-major. One wave per 16x16 C tile. K % 16 == 0; M % 16 == 0; NOUT ragged
// (clamped loads: an out-of-range B column only pollutes D columns that are
// never stored). grid = (ceil(NOUT/16), M/16), block = 32.
// ---------------------------------------------------------------------------
template <int K, int NOUT>
__global__ __launch_bounds__(32) void k_gemm_bias(
    const float* __restrict__ A, const float* __restrict__ W,
    const float* __restrict__ bias, float* __restrict__ C) {
  const int lane = threadIdx.x;
  const int nt = blockIdx.x;
  const long mt = blockIdx.y;
  const int ml = lane & 15;
  const int koff = (lane >> 4) * 2;
  const int n = nt * 16 + ml;
  const int nc = (n < NOUT) ? n : (NOUT - 1);  // clamped B column
  const float* __restrict__ ap = A + (mt * 16 + ml) * (long)K + koff;
  const float* __restrict__ wp = W + (long)koff * NOUT + nc;
  v8f acc = {};
#pragma unroll 4
  for (int k0 = 0; k0 < K; k0 += 4) {
    v2f a, b;
    a.x = ap[0];
    a.y = ap[1];
    b.x = wp[0];
    b.y = wp[NOUT];
    acc = wmma4(a, b, acc);
    ap += 4;
    wp += 4 * NOUT;
  }
  if (n < NOUT) {
    const float bv = bias[n];
    const int mbase = (lane >> 4) * 8;
#pragma unroll
    for (int r = 0; r < 8; ++r) {
      const long m = mt * 16 + mbase + r;
      C[m * NOUT + n] = acc[r] + bv;
    }
  }
}

// ---------------------------------------------------------------------------
// Rotation matrices / translations from quaternions; pw = sqrt(1/54)*softplus.
// ---------------------------------------------------------------------------
__global__ void k_prep(const float* __restrict__ recT,
                       const float* __restrict__ tw,
                       float* __restrict__ Rb, float* __restrict__ tb,
                       float* __restrict__ pw) {
  const int n = blockIdx.x * blockDim.x + threadIdx.x;
  if (n < NN) {
    float w = recT[n * 7 + 0], x = recT[n * 7 + 1];
    float y = recT[n * 7 + 2], z = recT[n * 7 + 3];
    const float inv = rsqrtf(w * w + x * x + y * y + z * z);
    w *= inv; x *= inv; y *= inv; z *= inv;
    Rb[n * 9 + 0] = 1.f - 2.f * (y * y + z * z);
    Rb[n * 9 + 1] = 2.f * (x * y - w * z);
    Rb[n * 9 + 2] = 2.f * (x * z + w * y);
    Rb[n * 9 + 3] = 2.f * (x * y + w * z);
    Rb[n * 9 + 4] = 1.f - 2.f * (x * x + z * z);
    Rb[n * 9 + 5] = 2.f * (y * z - w * x);
    Rb[n * 9 + 6] = 2.f * (x * z - w * y);
    Rb[n * 9 + 7] = 2.f * (y * z + w * x);
    Rb[n * 9 + 8] = 1.f - 2.f * (x * x + y * y);
    tb[n * 3 + 0] = recT[n * 7 + 4];
    tb[n * 3 + 1] = recT[n * 7 + 5];
    tb[n * 3 + 2] = recT[n * 7 + 6];
  }
  if (blockIdx.x == 0 && threadIdx.x < HH) {
    const float v = tw[threadIdx.x];
    const float sp = (v > 20.f) ? v : log1pf(__expf(v));
    pw[threadIdx.x] = 0.13608276348795434f * sp;  // sqrt(1/54)
  }
}

// ---------------------------------------------------------------------------
// Rotate local q/k/v points into global frame; precompute |q|^2, |k|^2.
// qpt/kpt: [H][N][12] (p*3+c), vpt: [H][N][24] (pv*3+c).
// ---------------------------------------------------------------------------
__global__ void k_points(const float* __restrict__ QP,
                         const float* __restrict__ KVP,
                         const float* __restrict__ Rb,
                         const float* __restrict__ tb,
                         float* __restrict__ qpt, float* __restrict__ kpt,
                         float* __restrict__ vpt, float* __restrict__ q2,
                         float* __restrict__ k2) {
  const int n = blockIdx.x * blockDim.x + threadIdx.x;
  if (n >= NN) return;
  float R[9], t[3];
#pragma unroll
  for (int i = 0; i < 9; ++i) R[i] = Rb[n * 9 + i];
#pragma unroll
  for (int i = 0; i < 3; ++i) t[i] = tb[n * 3 + i];

  for (int h = 0; h < HH; ++h) {
    float s2 = 0.f;
    for (int p = 0; p < 4; ++p) {
      const float l0 = QP[n * 144 + 0 * 48 + h * 4 + p];
      const float l1 = QP[n * 144 + 1 * 48 + h * 4 + p];
      const float l2 = QP[n * 144 + 2 * 48 + h * 4 + p];
#pragma unroll
      for (int c = 0; c < 3; ++c) {
        const float g = R[c * 3 + 0] * l0 + R[c * 3 + 1] * l1 +
                        R[c * 3 + 2] * l2 + t[c];
        qpt[((long)h * NN + n) * 12 + p * 3 + c] = g;
        s2 += g * g;
      }
    }
    q2[h * NN + n] = s2;
  }
  for (int h = 0; h < HH; ++h) {
    float s2 = 0.f;
    for (int d = 0; d < 12; ++d) {
      const float l0 = KVP[n * 432 + 0 * 144 + h * 12 + d];
      const float l1 = KVP[n * 432 + 1 * 144 + h * 12 + d];
      const float l2 = KVP[n * 432 + 2 * 144 + h * 12 + d];
#pragma unroll
      for (int c = 0; c < 3; ++c) {
        const float g = R[c * 3 + 0] * l0 + R[c * 3 + 1] * l1 +
                        R[c * 3 + 2] * l2 + t[c];
        if (d < 4) {
          kpt[((long)h * NN + n) * 12 + d * 3 + c] = g;
          s2 += g * g;
        } else {
          vpt[((long)h * NN + n) * 24 + (d - 4) * 3 + c] = g;
        }
      }
    }
    k2[h * NN + n] = s2;
  }
}

// ---------------------------------------------------------------------------
// Logits: scalar QK^T + point term + 2D bias, fused through WMMA.
// grid = (48 ktiles, 48 qtiles, 12 heads), block = 32.
// ---------------------------------------------------------------------------
__global__ __launch_bounds__(32) void k_logits(
    const float* __restrict__ Q, const float* __restrict__ KV,
    const float* __restrict__ qpt, const float* __restrict__ kpt,
    const float* __restrict__ q2, const float* __restrict__ k2,
    const float* __restrict__ Z, const float* __restrict__ pw,
    float* __restrict__ L) {
  const int lane = threadIdx.x;
  const int kt = blockIdx.x, qt = blockIdx.y, h = blockIdx.z;
  const int ml = lane & 15;
  const int koff = (lane >> 4) * 2;
  const long qrow = qt * 16 + ml;        // A-side row (query)
  const long krow = (long)kt * 16 + ml;  // B-side column (key)
  const float sw = 0.14433756729740643f; // sqrt(1/(3*16))
  const float pwh = pw[h];
  v8f acc = {};
  // scalar term: K = SQK = 16 (A pre-scaled by sw)
  {
    const float* __restrict__ ap = Q + qrow * 192 + h * 16 + koff;
    const float* __restrict__ bp = KV + krow * 384 + h * 32 + koff;
#pragma unroll
    for (int k0 = 0; k0 < 16; k0 += 4) {
      v2f a, b;
      a.x = sw * ap[0];
      a.y = sw * ap[1];
      b.x = bp[0];
      b.y = bp[1];
      acc = wmma4(a, b, acc);
      ap += 4;
      bp += 4;
    }
  }
  // point cross term: K = PQK*3 = 12 (A pre-scaled by pw[h])
  {
    const float* __restrict__ ap = qpt + ((long)h * NN + qrow) * 12 + koff;
    const float* __restrict__ bp = kpt + ((long)h * NN + krow) * 12 + koff;
#pragma unroll
    for (int k0 = 0; k0 < 12; k0 += 4) {
      v2f a, b;
      a.x = pwh * ap[0];
      a.y = pwh * ap[1];
      b.x = bp[0];
      b.y = bp[1];
      acc = wmma4(a, b, acc);
      ap += 4;
      bp += 4;
    }
  }
  const float r3 = 0.5773502691896258f;  // sqrt(1/3)
  const int j = kt * 16 + ml;
  const int mbase = (lane >> 4) * 8;
#pragma unroll
  for (int r = 0; r < 8; ++r) {
    const int qi = qt * 16 + mbase + r;
    const float val = acc[r] - 0.5f * pwh * (q2[h * NN + qi] + k2[h * NN + j]) +
                      r3 * Z[((long)qi * NN + j) * HH + h];
    L[((long)h * NN + qi) * NN + j] = val;
  }
}

// ---------------------------------------------------------------------------
// Row softmax over k (rows of length 768 = 3*256). In-place.
// ---------------------------------------------------------------------------
__global__ __launch_bounds__(256) void k_softmax(float* __restrict__ L) {
  const long row = blockIdx.x;
  float* p = L + row * NN;
  __shared__ float red[256];
  const int t = threadIdx.x;
  float m = -1e30f;
#pragma unroll
  for (int i = 0; i < 3; ++i) m = fmaxf(m, p[t + i * 256]);
  red[t] = m; __syncthreads();
  for (int s = 128; s > 0; s >>= 1) {
    if (t < s) red[t] = fmaxf(red[t], red[t + s]);
    __syncthreads();
  }
  m = red[0]; __syncthreads();
  float vals[3]; float sum = 0.f;
#pragma unroll
  for (int i = 0; i < 3; ++i) { vals[i] = __expf(p[t + i * 256] - m); sum += vals[i]; }
  red[t] = sum; __syncthreads();
  for (int s = 128; s > 0; s >>= 1) {
    if (t < s) red[t] += red[t + s];
    __syncthreads();
  }
  const float inv = 1.f / red[0];
#pragma unroll
  for (int i = 0; i < 3; ++i) p[t + i * 256] = vals[i] * inv;
}

// ---------------------------------------------------------------------------
// result_scalar = attn @ v_scalar -> FA[:, h*16+s]. grid=(48 qt, 12 h).
// ---------------------------------------------------------------------------
__global__ __launch_bounds__(32) void k_res_scalar(
    const float* __restrict__ attn, const float* __restrict__ KV,
    float* __restrict__ FA) {
  const int lane = threadIdx.x;
  const int qt = blockIdx.x, h = blockIdx.y;
  const int ml = lane & 15;
  const int koff = (lane >> 4) * 2;
  const float* __restrict__ ap =
      attn + (long)h * NN * NN + (long)(qt * 16 + ml) * NN + koff;
  const float* __restrict__ bp = KV + (long)koff * 384 + h * 32 + 16 + ml;
  v8f acc = {};
#pragma unroll 4
  for (int k0 = 0; k0 < NN; k0 += 4) {
    v2f a, b;
    a.x = ap[0];
    a.y = ap[1];
    b.x = bp[0];
    b.y = bp[384];
    acc = wmma4(a, b, acc);
    ap += 4;
    bp += 4 * 384;
  }
  const int mbase = (lane >> 4) * 8;
#pragma unroll
  for (int r = 0; r < 8; ++r) {
    const int q = qt * 16 + mbase + r;
    FA[(long)q * 2112 + h * 16 + ml] = acc[r];
  }
}

// ---------------------------------------------------------------------------
// res_pt = attn @ v_pt (N=24, clamped loads). grid=(2 nt, 48 qt, 12 h).
// ---------------------------------------------------------------------------
__global__ __launch_bounds__(32) void k_res_pt(
    const float* __restrict__ attn, const float* __restrict__ vpt,
    float* __restrict__ rpt) {
  const int lane = threadIdx.x;
  const int nt = blockIdx.x, qt = blockIdx.y, h = blockIdx.z;
  const int ml = lane & 15;
  const int koff = (lane >> 4) * 2;
  const int n = nt * 16 + ml;
  const int nc = (n < 24) ? n : 23;  // clamped B column
  const float* __restrict__ ap =
      attn + (long)h * NN * NN + (long)(qt * 16 + ml) * NN + koff;
  const float* __restrict__ bp = vpt + (long)h * NN * 24 + (long)koff * 24 + nc;
  v8f acc = {};
#pragma unroll 4
  for (int k0 = 0; k0 < NN; k0 += 4) {
    v2f a, b;
    a.x = ap[0];
    a.y = ap[1];
    b.x = bp[0];
    b.y = bp[24];
    acc = wmma4(a, b, acc);
    ap += 4;
    bp += 4 * 24;
  }
  if (n < 24) {
    const int mbase = (lane >> 4) * 8;
#pragma unroll
    for (int r = 0; r < 8; ++r) {
      const int q = qt * 16 + mbase + r;
      rpt[((long)h * NN + q) * 24 + n] = acc[r];
    }
  }
}

// ---------------------------------------------------------------------------
// att2d[i,h,c] = sum_j attn[h,i,j]*rep2d[i,j,c]: per query i a
// [12(pad16), 768] @ [768, 128] GEMM. grid=(8 ctiles, 768 queries).
// A row (head) clamped: rows >= 12 pollute only unstored D rows.
// ---------------------------------------------------------------------------
__global__ __launch_bounds__(32) void k_att2d(
    const float* __restrict__ attn, const float* __restrict__ rep2d,
    float* __restrict__ FA) {
  const int lane = threadIdx.x;
  const int ct = blockIdx.x;
  const long i = blockIdx.y;
  const int ml = lane & 15;
  const int koff = (lane >> 4) * 2;
  const int hc = (ml < HH) ? ml : (HH - 1);  // clamped A row (head)
  const float* __restrict__ ap = attn + ((long)hc * NN + i) * NN + koff;
  const float* __restrict__ bp =
      rep2d + i * (long)NN * 128 + (long)koff * 128 + ct * 16 + ml;
  v8f acc = {};
#pragma unroll 4
  for (int k0 = 0; k0 < NN; k0 += 4) {
    v2f a, b;
    a.x = ap[0];
    a.y = ap[1];
    b.x = bp[0];
    b.y = bp[128];
    __builtin_prefetch(bp + 16 * 128, 0, 1);
    acc = wmma4(a, b, acc);
    ap += 4;
    bp += 4 * 128;
  }
  const int c = ct * 16 + ml;
  const int mbase = (lane >> 4) * 8;
#pragma unroll
  for (int r = 0; r < 8; ++r) {
    const int hh = mbase + r;
    if (hh < HH) FA[i * 2112 + 576 + (long)hh * 128 + c] = acc[r];
  }
}

// ---------------------------------------------------------------------------
// local coords (inverse rotation) + pt_norm into FA.
// ---------------------------------------------------------------------------
__global__ void k_local(const float* __restrict__ rpt,
                        const float* __restrict__ Rb,
                        const float* __restrict__ tb,
                        float* __restrict__ FA) {
  const int idx = blockIdx.x * blockDim.x + threadIdx.x;
  if (idx >= NN * 96) return;
  const int n = idx / 96;
  const int d = idx % 96;  // d = h*8 + pv
  const int h = d >> 3;
  const int pv = d & 7;
  float g0 = rpt[((long)h * NN + n) * 24 + pv * 3 + 0] - tb[n * 3 + 0];
  float g1 = rpt[((long)h * NN + n) * 24 + pv * 3 + 1] - tb[n * 3 + 1];
  float g2 = rpt[((long)h * NN + n) * 24 + pv * 3 + 2] - tb[n * 3 + 2];
  float s2 = 1e-8f;
  float* fa = FA + (long)n * 2112;
#pragma unroll
  for (int i = 0; i < 3; ++i) {
    // local[n,i,d] = sum_c R[n,c,i] * (g[c] - t[c])   (R^T)
    const float li = Rb[n * 9 + 0 * 3 + i] * g0 + Rb[n * 9 + 1 * 3 + i] * g1 +
                     Rb[n * 9 + 2 * 3 + i] * g2;
    fa[192 + i * 96 + d] = li;
    s2 += li * li;
  }
  fa[480 + d] = sqrtf(s2);
}

// ---------------------------------------------------------------------------
// Host-side launcher.
// ---------------------------------------------------------------------------
extern "C" void kernel_launch(void* const* d_in, const int* in_sizes, int n_in,
                              void* d_out, int out_size, void* d_ws,
                              size_t ws_size, hipStream_t stream) {
  (void)in_sizes; (void)n_in; (void)out_size; (void)ws_size;
  const float* rec1d = (const float*)d_in[0];
  const float* rep2d = (const float*)d_in[1];
  const float* recT  = (const float*)d_in[2];
  const float* q_w   = (const float*)d_in[3];
  const float* q_b   = (const float*)d_in[4];
  const float* kv_w  = (const float*)d_in[5];
  const float* kv_b  = (const float*)d_in[6];
  const float* qp_w  = (const float*)d_in[7];
  const float* qp_b  = (const float*)d_in[8];
  const float* kvp_w = (const float*)d_in[9];
  const float* kvp_b = (const float*)d_in[10];
  const float* w2d_w = (const float*)d_in[11];
  const float* w2d_b = (const float*)d_in[12];
  const float* fin_w = (const float*)d_in[13];
  const float* fin_b = (const float*)d_in[14];
  const float* tw    = (const float*)d_in[15];
  float* out = (float*)d_out;

  float* ws = (float*)d_ws;
  float* Rb  = ws; ws += NN * 9;
  float* tb  = ws; ws += NN * 3;
  float* pw  = ws; ws += 16;
  float* Q   = ws; ws += NN * 192;        // q_scalar
  float* KV  = ws; ws += NN * 384;        // k_scalar|v_scalar
  float* QP  = ws; ws += NN * 144;
  float* KVP = ws; ws += NN * 432;
  float* qpt = ws; ws += (long)HH * NN * 12;
  float* kpt = ws; ws += (long)HH * NN * 12;
  float* vpt = ws; ws += (long)HH * NN * 24;
  float* q2  = ws; ws += HH * NN;
  float* k2  = ws; ws += HH * NN;
  float* Z   = ws; ws += (long)NN * NN * HH;  // attn_2d projection
  float* L   = ws; ws += (long)NN * NN * HH;  // logits -> attn (in place)
  float* rpt = ws; ws += (long)HH * NN * 24;
  float* FA  = ws; ws += (long)NN * 2112;     // concat buffer

  // 1) rotations / translations / point weights
  k_prep<<<dim3(3), dim3(256), 0, stream>>>(recT, tw, Rb, tb, pw);

  // 2) linear projections (WMMA f32 GEMMs)
  k_gemm_bias<384, 192><<<dim3(12, 48), dim3(32), 0, stream>>>(rec1d, q_w, q_b, Q);
  k_gemm_bias<384, 384><<<dim3(24, 48), dim3(32), 0, stream>>>(rec1d, kv_w, kv_b, KV);
  k_gemm_bias<384, 144><<<dim3(9, 48), dim3(32), 0, stream>>>(rec1d, qp_w, qp_b, QP);
  k_gemm_bias<384, 432><<<dim3(27, 48), dim3(32), 0, stream>>>(rec1d, kvp_w, kvp_b, KVP);

  // 3) 2D bias projection: [768*768, 128] @ [128, 12]
  k_gemm_bias<128, HH><<<dim3(1, (NN * NN) / 16), dim3(32), 0, stream>>>(
      rep2d, w2d_w, w2d_b, Z);

  // 4) point transforms
  k_points<<<dim3(3), dim3(256), 0, stream>>>(QP, KVP, Rb, tb, qpt, kpt, vpt, q2, k2);

  // 5) fused attention logits (WMMA)
  k_logits<<<dim3(48, 48, HH), dim3(32), 0, stream>>>(Q, KV, qpt, kpt, q2, k2, Z, pw, L);

  // 6) softmax over keys
  k_softmax<<<dim3(HH * NN), dim3(256), 0, stream>>>(L);

  // 7) attention outputs (WMMA)
  k_res_scalar<<<dim3(48, HH), dim3(32), 0, stream>>>(L, KV, FA);
  k_res_pt<<<dim3(2, 48, HH), dim3(32), 0, stream>>>(L, vpt, rpt);
  k_att2d<<<dim3(8, NN), dim3(32), 0, stream>>>(L, rep2d, FA);

  // 8) inverse rotation + norms
  k_local<<<dim3(288), dim3(256), 0, stream>>>(rpt, Rb, tb, FA);

  // 9) final projection: [768, 2112] @ [2112, 384] -> out
  k_gemm_bias<2112, 384><<<dim3(24, 48), dim3(32), 0, stream>>>(FA, fin_w, fin_b, out);
}